// BayesMultiHeadAttn_19619410608619
// MI455X (gfx1250) — compile-verified
//
#include <hip/hip_runtime.h>
#include <hip/hip_bf16.h>

#define SEQ    1024
#define BATCH  8
#define DMODEL 1024
#define NHEAD  16
#define DHEAD  64
#define ROWS   (SEQ * BATCH)      // 8192
#define SCALEF 0.125f             // 1/sqrt(64)
#define LN_EPS 1e-5f

typedef float v2f __attribute__((ext_vector_type(2)));
typedef float v8f __attribute__((ext_vector_type(8)));

// ---------------------------------------------------------------------------
// Fragment load for V_WMMA_F32_16X16X4_F32.
// A (16x4, MxK): lanes 0-15 hold M=lane / K=0,1 ; lanes 16-31 hold M=lane-16 / K=2,3.
// B (4x16, KxN): lanes 0-15 hold N=lane / K=0,1 ; lanes 16-31 hold N=lane-16 / K=2,3.
// Both are a contiguous float2 when K is the fastest-moving dim (global_load_b64).
// ---------------------------------------------------------------------------
__device__ __forceinline__ v2f frag_ld(const float* __restrict__ p, int row0,
                                       int ld, int k0, int lane) {
  const float* q = p + (size_t)(row0 + (lane & 15)) * ld + (k0 + 2 * (lane >> 4));
  v2f f; f.x = q[0]; f.y = q[1];
  return f;
}

__device__ __forceinline__ v8f wmma4(v2f a, v2f b, v8f c) {
  return __builtin_amdgcn_wmma_f32_16x16x4_f32(false, a, false, b, (short)0, c,
                                               false, false);
}

// ---------------------------------------------------------------------------
// 1) Bayesian weight sampling: w = mean + eps * exp(std).
//    (reference bug preserved: ow uses value_mean, not output_mean)
// ---------------------------------------------------------------------------
__global__ __launch_bounds__(256) void sample_weights_kernel(
    const float* __restrict__ qm, const float* __restrict__ km,
    const float* __restrict__ vm, const float* __restrict__ qs,
    const float* __restrict__ ks, const float* __restrict__ vs,
    const float* __restrict__ os_, const float* __restrict__ eq,
    const float* __restrict__ ek, const float* __restrict__ ev,
    const float* __restrict__ eo, float* __restrict__ qw,
    float* __restrict__ kw, float* __restrict__ vw, float* __restrict__ ow) {
  int i = blockIdx.x * 256 + threadIdx.x;  // grid covers 1024*1024 exactly
  float vmi = vm[i];
  qw[i] = qm[i] + eq[i] * __expf(qs[i]);
  kw[i] = km[i] + ek[i] * __expf(ks[i]);
  vw[i] = vmi   + ev[i] * __expf(vs[i]);
  ow[i] = vmi   + eo[i] * __expf(os_[i]);   // faithful to reference bug
}

// ---------------------------------------------------------------------------
// 2) QKV projections: {Q,K,V}[r,o] = sum_d A[r,d] * W{q,k,v}[o,d]
//    One wave -> one 16(M)x32(N) tile for all three outputs.
//    A fragment reused by 6 WMMAs; next-k fragments double-buffered in regs.
//    grid = (ROWS/16, DMODEL/128), block = 128 (4 waves along o).
// ---------------------------------------------------------------------------
__global__ __launch_bounds__(128) void qkv_gemm_kernel(
    const float* __restrict__ A, const float* __restrict__ qw,
    const float* __restrict__ kw, const float* __restrict__ vw,
    float* __restrict__ Q, float* __restrict__ K, float* __restrict__ V) {
  const int lane = threadIdx.x & 31;
  const int wave = threadIdx.x >> 5;
  const int r0 = blockIdx.x * 16;
  const int o0 = (blockIdx.y * 4 + wave) * 32;

  v8f cq0 = {}, cq1 = {}, ck0 = {}, ck1 = {}, cv0 = {}, cv1 = {};

  // prologue: fragments for k-step 0
  v2f a  = frag_ld(A,  r0,      DMODEL, 0, lane);
  v2f q0 = frag_ld(qw, o0,      DMODEL, 0, lane);
  v2f q1 = frag_ld(qw, o0 + 16, DMODEL, 0, lane);
  v2f k0 = frag_ld(kw, o0,      DMODEL, 0, lane);
  v2f k1 = frag_ld(kw, o0 + 16, DMODEL, 0, lane);
  v2f v0 = frag_ld(vw, o0,      DMODEL, 0, lane);
  v2f v1 = frag_ld(vw, o0 + 16, DMODEL, 0, lane);

#pragma unroll 8
  for (int d0 = 0; d0 < DMODEL - 4; d0 += 4) {
    const int dn = d0 + 4;
    // prefetch next k-step into a second register set (covers load latency
    // with the 6 WMMAs below instead of stalling on s_wait_loadcnt)
    v2f an  = frag_ld(A,  r0,      DMODEL, dn, lane);
    v2f qn0 = frag_ld(qw, o0,      DMODEL, dn, lane);
    v2f qn1 = frag_ld(qw, o0 + 16, DMODEL, dn, lane);
    v2f kn0 = frag_ld(kw, o0,      DMODEL, dn, lane);
    v2f kn1 = frag_ld(kw, o0 + 16, DMODEL, dn, lane);
    v2f vn0 = frag_ld(vw, o0,      DMODEL, dn, lane);
    v2f vn1 = frag_ld(vw, o0 + 16, DMODEL, dn, lane);

    cq0 = wmma4(a, q0, cq0);
    cq1 = wmma4(a, q1, cq1);
    ck0 = wmma4(a, k0, ck0);
    ck1 = wmma4(a, k1, ck1);
    cv0 = wmma4(a, v0, cv0);
    cv1 = wmma4(a, v1, cv1);

    a = an; q0 = qn0; q1 = qn1; k0 = kn0; k1 = kn1; v0 = vn0; v1 = vn1;
  }
  // epilogue k-step
  cq0 = wmma4(a, q0, cq0);
  cq1 = wmma4(a, q1, cq1);
  ck0 = wmma4(a, k0, ck0);
  ck1 = wmma4(a, k1, ck1);
  cv0 = wmma4(a, v0, cv0);
  cv1 = wmma4(a, v1, cv1);

  const int n = lane & 15;
  const int mb = (lane >> 4) * 8;
#pragma unroll
  for (int i = 0; i < 8; ++i) {
    size_t idx = (size_t)(r0 + mb + i) * DMODEL + o0 + n;
    Q[idx] = cq0[i];
    K[idx] = ck0[i];
    V[idx] = cv0[i];
    Q[idx + 16] = cq1[i];
    K[idx + 16] = ck1[i];
    V[idx + 16] = cv1[i];
  }
}

// ---------------------------------------------------------------------------
// 3) Causal flash attention. One wave per (batch, head, 16-query tile).
//    grid = (B*H*(SEQ/16))/4 blocks, block = 128.
// ---------------------------------------------------------------------------
__global__ __launch_bounds__(128) void attn_kernel(
    const float* __restrict__ Q, const float* __restrict__ K,
    const float* __restrict__ V, float* __restrict__ O) {
  const int lane = threadIdx.x & 31;
  const int wave = threadIdx.x >> 5;
  const int gw = blockIdx.x * 4 + wave;   // 0 .. 8191
  const int it = gw & 63;                 // query tile
  const int h  = (gw >> 6) & 15;          // head
  const int b  = gw >> 10;                // batch
  const int i0 = it * 16;
  const int hc = h * DHEAD;
  const int half = lane >> 4;
  const int ln = lane & 15;
  const size_t rstride = (size_t)BATCH * DMODEL;  // seq-row stride

  __shared__ float lds[4][16][17];
  float (*P)[17] = lds[wave];

  float mrow[8], lrow[8];
  v8f acc[4] = {v8f{}, v8f{}, v8f{}, v8f{}};
#pragma unroll
  for (int i = 0; i < 8; ++i) { mrow[i] = -3.0e38f; lrow[i] = 0.0f; }

  const float* Qb = Q + (size_t)b * DMODEL + hc;
  const float* Kb = K + (size_t)b * DMODEL + hc;
  const float* Vb = V + (size_t)b * DMODEL + hc;

  for (int j0 = 0; j0 <= i0; j0 += 16) {
    // ---- S = (Q Kt) over d = 0..63 ----
    v8f s8 = {};
#pragma unroll
    for (int kk = 0; kk < DHEAD; kk += 4) {
      v2f a, bb;
      const float* qa = Qb + (size_t)(i0 + ln) * rstride + kk + 2 * half;
      a.x = qa[0]; a.y = qa[1];
      const float* kb = Kb + (size_t)(j0 + ln) * rstride + kk + 2 * half;
      bb.x = kb[0]; bb.y = kb[1];
      s8 = wmma4(a, bb, s8);
    }
    // ---- scale + causal mask (element (m,n): q = i0+m, k = j0+n) ----
    float sv[8];
#pragma unroll
    for (int i = 0; i < 8; ++i) {
      int m = i + 8 * half;
      sv[i] = (j0 + ln > i0 + m) ? -3.0e38f : s8[i] * SCALEF;
    }
    // ---- row max across the 16 lanes holding each row ----
    float rmax[8];
#pragma unroll
    for (int i = 0; i < 8; ++i) {
      float v = sv[i];
      for (int msk = 1; msk < 16; msk <<= 1) v = fmaxf(v, __shfl_xor(v, msk, 16));
      rmax[i] = v;
    }
    float alpha[8];
#pragma unroll
    for (int i = 0; i < 8; ++i) {
      float mn = fmaxf(mrow[i], rmax[i]);
      alpha[i] = __expf(mrow[i] - mn);
      mrow[i] = mn;
    }
    // ---- P = exp(S - m), row sums, rescale state ----
    float pv[8];
#pragma unroll
    for (int i = 0; i < 8; ++i) pv[i] = __expf(sv[i] - mrow[i]);
#pragma unroll
    for (int i = 0; i < 8; ++i) {
      float v = pv[i];
      for (int msk = 1; msk < 16; msk <<= 1) v += __shfl_xor(v, msk, 16);
      lrow[i] = lrow[i] * alpha[i] + v;
    }
#pragma unroll
    for (int t = 0; t < 4; ++t)
#pragma unroll
      for (int i = 0; i < 8; ++i) acc[t][i] *= alpha[i];
    // ---- reshape P (C layout -> A layout) through per-wave LDS tile ----
#pragma unroll
    for (int i = 0; i < 8; ++i) P[i + 8 * half][ln] = pv[i];
    // same-wave DS ordering: compiler inserts s_wait_dscnt before the reads
    // ---- acc += P @ V  (K = 16 over j, N = 64 over d in 4 tiles) ----
#pragma unroll
    for (int t = 0; t < 4; ++t) {
      int n0 = t * 16;
#pragma unroll
      for (int kk = 0; kk < 16; kk += 4) {
        v2f a, bb;
        a.x = P[ln][kk + 2 * half];
        a.y = P[ln][kk + 2 * half + 1];
        const float* vp = Vb + (size_t)(j0 + kk + 2 * half) * rstride + n0 + ln;
        bb.x = vp[0];
        bb.y = vp[rstride];   // next j row
        acc[t] = wmma4(a, bb, acc[t]);
      }
    }
  }
  // ---- O = acc / l ----
#pragma unroll
  for (int t = 0; t < 4; ++t)
#pragma unroll
    for (int i = 0; i < 8; ++i) {
      int m = i + 8 * half;
      O[(size_t)(i0 + m) * rstride + (size_t)b * DMODEL + hc + t * 16 + ln] =
          acc[t][i] / lrow[i];
    }
}

// ---------------------------------------------------------------------------
// 4) Output projection + residual: X[r,o] = dec[r,o] + sum_d Avec[r,d]*ow[o,d]
//    16x32 tile per wave, double-buffered fragments.
// ---------------------------------------------------------------------------
__global__ __launch_bounds__(128) void oproj_kernel(
    const float* __restrict__ Avec, const float* __restrict__ ow,
    const float* __restrict__ resid, float* __restrict__ X) {
  const int lane = threadIdx.x & 31;
  const int wave = threadIdx.x >> 5;
  const int r0 = blockIdx.x * 16;
  const int o0 = (blockIdx.y * 4 + wave) * 32;

  v8f c0 = {}, c1 = {};
  v2f a  = frag_ld(Avec, r0,      DMODEL, 0, lane);
  v2f b0 = frag_ld(ow,   o0,      DMODEL, 0, lane);
  v2f b1 = frag_ld(ow,   o0 + 16, DMODEL, 0, lane);

#pragma unroll 8
  for (int d0 = 0; d0 < DMODEL - 4; d0 += 4) {
    const int dn = d0 + 4;
    v2f an  = frag_ld(Avec, r0,      DMODEL, dn, lane);
    v2f bn0 = frag_ld(ow,   o0,      DMODEL, dn, lane);
    v2f bn1 = frag_ld(ow,   o0 + 16, DMODEL, dn, lane);
    c0 = wmma4(a, b0, c0);
    c1 = wmma4(a, b1, c1);
    a = an; b0 = bn0; b1 = bn1;
  }
  c0 = wmma4(a, b0, c0);
  c1 = wmma4(a, b1, c1);

  const int n = lane & 15;
  const int mb = (lane >> 4) * 8;
#pragma unroll
  for (int i = 0; i < 8; ++i) {
    size_t idx = (size_t)(r0 + mb + i) * DMODEL + o0 + n;
    X[idx] = c0[i] + resid[idx];
    X[idx + 16] = c1[i] + resid[idx + 16];
  }
}

// ---------------------------------------------------------------------------
// 5) LayerNorm per row (1024 elems, 256 threads x 4).
// ---------------------------------------------------------------------------
__global__ __launch_bounds__(256) void layernorm_kernel(
    const float* __restrict__ X, const float* __restrict__ gamma,
    const float* __restrict__ beta, float* __restrict__ out) {
  const int row = blockIdx.x;
  const float* x = X + (size_t)row * DMODEL;
  float v[4], s = 0.f, ss = 0.f;
#pragma unroll
  for (int k = 0; k < 4; ++k) {
    v[k] = x[threadIdx.x + k * 256];
    s += v[k];
    ss += v[k] * v[k];
  }
  for (int msk = 1; msk < 32; msk <<= 1) {
    s += __shfl_xor(s, msk, 32);
    ss += __shfl_xor(ss, msk, 32);
  }
  __shared__ float sb[8], ssb[8];
  const int wave = threadIdx.x >> 5;
  if ((threadIdx.x & 31) == 0) { sb[wave] = s; ssb[wave] = ss; }
  __syncthreads();
  s = 0.f; ss = 0.f;
#pragma unroll
  for (int w = 0; w < 8; ++w) { s += sb[w]; ss += ssb[w]; }
  const float mu = s * (1.0f / DMODEL);
  const float var = ss * (1.0f / DMODEL) - mu * mu;
  const float r = rsqrtf(var + LN_EPS);
#pragma unroll
  for (int k = 0; k < 4; ++k) {
    int cidx = threadIdx.x + k * 256;
    out[(size_t)row * DMODEL + cidx] = (v[k] - mu) * r * gamma[cidx] + beta[cidx];
  }
}

// ---------------------------------------------------------------------------
// 6) KL: mean over 4 x (1024x1024) of (m^2 - 2s + exp(2s)) / 2.
//    Two-stage deterministic reduction (graph-replay bit-stable).
// ---------------------------------------------------------------------------
__global__ __launch_bounds__(256) void kl_stage1_kernel(
    const float* __restrict__ qm, const float* __restrict__ km,
    const float* __restrict__ vm, const float* __restrict__ om,
    const float* __restrict__ qs, const float* __restrict__ ks,
    const float* __restrict__ vs, const float* __restrict__ os_,
    float* __restrict__ partial) {
  float acc = 0.f;
  for (int i = blockIdx.x * 256 + threadIdx.x; i < DMODEL * DMODEL;
       i += gridDim.x * 256) {
    acc += qm[i] * qm[i] - 2.f * qs[i] + __expf(2.f * qs[i]);
    acc += km[i] * km[i] - 2.f * ks[i] + __expf(2.f * ks[i]);
    acc += vm[i] * vm[i] - 2.f * vs[i] + __expf(2.f * vs[i]);
    acc += om[i] * om[i] - 2.f * os_[i] + __expf(2.f * os_[i]);
  }
  for (int msk = 1; msk < 32; msk <<= 1) acc += __shfl_xor(acc, msk, 32);
  __shared__ float sb[8];
  if ((threadIdx.x & 31) == 0) sb[threadIdx.x >> 5] = acc;
  __syncthreads();
  if (threadIdx.x == 0) {
    float t = 0.f;
#pragma unroll
    for (int w = 0; w < 8; ++w) t += sb[w];
    partial[blockIdx.x] = t;
  }
}

__global__ __launch_bounds__(256) void kl_stage2_kernel(
    const float* __restrict__ partial, float* __restrict__ kl_out) {
  float acc = partial[threadIdx.x] + partial[threadIdx.x + 256];
  for (int msk = 1; msk < 32; msk <<= 1) acc += __shfl_xor(acc, msk, 32);
  __shared__ float sb[8];
  if ((threadIdx.x & 31) == 0) sb[threadIdx.x >> 5] = acc;
  __syncthreads();
  if (threadIdx.x == 0) {
    float t = 0.f;
#pragma unroll
    for (int w = 0; w < 8; ++w) t += sb[w];
    // mean over 4*1024*1024 elements, then / 2
    kl_out[0] = t * (0.5f / 4194304.0f);
  }
}

// ---------------------------------------------------------------------------
extern "C" void kernel_launch(void* const* d_in, const int* in_sizes, int n_in,
                              void* d_out, int out_size, void* d_ws,
                              size_t ws_size, hipStream_t stream) {
  const float* dec = (const float*)d_in[0];
  // d_in[1] = attn_mask (causal triu) — applied analytically, not read.
  const float* qm = (const float*)d_in[2];
  const float* km = (const float*)d_in[3];
  const float* vm = (const float*)d_in[4];
  const float* qs = (const float*)d_in[5];
  const float* ks = (const float*)d_in[6];
  const float* vs = (const float*)d_in[7];
  const float* om = (const float*)d_in[8];
  const float* os_ = (const float*)d_in[9];
  const float* gamma = (const float*)d_in[10];
  const float* beta = (const float*)d_in[11];
  const float* eq = (const float*)d_in[12];
  const float* ek = (const float*)d_in[13];
  const float* ev = (const float*)d_in[14];
  const float* eo = (const float*)d_in[15];

  float* ws = (float*)d_ws;
  const size_t MB1 = (size_t)DMODEL * DMODEL;  // 1M floats
  const size_t RD  = (size_t)ROWS * DMODEL;    // 8M floats
  float* qw = ws;
  float* kw = ws + MB1;
  float* vw = ws + 2 * MB1;
  float* ow = ws + 3 * MB1;
  float* Q  = ws + 4 * MB1;
  float* K  = Q + RD;
  float* V  = K + RD;
  float* Ov = V + RD;
  float* X  = Q;                 // Q retired after attention; reuse for residual sum
  float* kl_part = kw;           // kw retired after QKV GEMM; 512 floats

  float* out = (float*)d_out;
  float* kl_out = out + RD;      // out_size = ROWS*DMODEL + 1

  sample_weights_kernel<<<(int)(MB1 / 256), 256, 0, stream>>>(
      qm, km, vm, qs, ks, vs, os_, eq, ek, ev, eo, qw, kw, vw, ow);

  qkv_gemm_kernel<<<dim3(ROWS / 16, DMODEL / 128), 128, 0, stream>>>(
      dec, qw, kw, vw, Q, K, V);

  attn_kernel<<<(BATCH * NHEAD * (SEQ / 16)) / 4, 128, 0, stream>>>(Q, K, V, Ov);

  oproj_kernel<<<dim3(ROWS / 16, DMODEL / 128), 128, 0, stream>>>(Ov, ow, dec, X);

  layernorm_kernel<<<ROWS, 256, 0, stream>>>(X, gamma, beta, out);

  kl_stage1_kernel<<<512, 256, 0, stream>>>(qm, km, vm, om, qs, ks, vs, os_,
                                            kl_part);
  kl_stage2_kernel<<<1, 256, 0, stream>>>(kl_part, kl_out);
}